// PfaffianState_37185826848961
// MI455X (gfx1250) — compile-verified
//
#include <hip/hip_runtime.h>
#include <hip/hip_bf16.h>
#include <math.h>

typedef __attribute__((ext_vector_type(2))) float v2f;
typedef __attribute__((ext_vector_type(8))) float v8f;

#define BATCH_N  1024
#define NE       64     // matrix dimension (N_ELEC)
#define NTHREADS 128    // 4 wave32
#define NB       16     // LU block size == WMMA tile

__device__ __forceinline__ float2 cmulf(float2 a, float2 b) {
    return make_float2(a.x * b.x - a.y * b.y, a.x * b.y + a.y * b.x);
}

__global__ __launch_bounds__(NTHREADS)
void pfaffian_logdet_kernel(const long long* __restrict__ yg,
                            const double* __restrict__ Fg,
                            double* __restrict__ outg)
{
    __shared__ float2  Ms[NE][NE + 1];      // fp32 complex working matrix (~33 KB)
    __shared__ double2 stage[16 * NE];      // async-gather staging, 16 rows (16 KB)
    __shared__ int     sy[NE];
    __shared__ double  s_lre, s_lim;
    __shared__ int     s_piv, s_nsw;
    __shared__ float2  s_inv;

    const int tid = threadIdx.x;
    const int b   = blockIdx.x;

    if (tid < NE) sy[tid] = (int)yg[(size_t)b * NE + tid];
    if (tid == 0) { s_lre = 0.0; s_lim = 0.0; s_nsw = 0; }
    __syncthreads();

    const char* fb = (const char*)Fg;

    // ---- Gather F[y_i, y_j] (complex128) via async global->LDS, 4 chunks of 16 rows,
    //      convert to fp32 complex into Ms. F table is L2-resident (64MB < 192MB L2).
    for (int c = 0; c < 4; ++c) {
        #pragma unroll
        for (int it = 0; it < 8; ++it) {
            const int e  = tid + NTHREADS * it;       // 0..1023 (16 rows x 64 cols)
            const int il = e >> 6, j = e & 63;
            const int i  = c * 16 + il;
            unsigned ldsoff = (unsigned)(uintptr_t)(&stage[e]);
            const char* gp = fb + ((((size_t)sy[i] << 11) + (size_t)sy[j]) << 4);
            asm volatile("global_load_async_to_lds_b128 %0, %1, off"
                         :: "v"(ldsoff), "v"(gp) : "memory");
        }
        asm volatile("s_wait_asynccnt 0" ::: "memory");
        __syncthreads();
        for (int it = 0; it < 8; ++it) {
            const int e  = tid + NTHREADS * it;
            const int il = e >> 6, j = e & 63;
            const int i  = c * 16 + il;
            double2 v = stage[e];
            Ms[i][j] = make_float2((float)v.x, (float)v.y);
        }
        __syncthreads();
    }

    // ---- Skew-symmetrize in place: M = F_occ - F_occ^T
    for (int idx = tid; idx < NE * NE; idx += NTHREADS) {
        const int i = idx >> 6, j = idx & 63;
        if (i < j) {
            float2 a = Ms[i][j], bb = Ms[j][i];
            float2 t = make_float2(a.x - bb.x, a.y - bb.y);
            Ms[i][j] = t;
            Ms[j][i] = make_float2(-t.x, -t.y);
        } else if (i == j) {
            Ms[i][i] = make_float2(0.f, 0.f);
        }
    }
    __syncthreads();

    // ---- Blocked complex LU with partial pivoting; logdet accumulated in f64.
    for (int kb = 0; kb < 4; ++kb) {
        const int kbase = kb * NB;
        for (int kk = 0; kk < NB; ++kk) {
            const int k = kbase + kk;
            if (tid == 0) {
                int piv = k; float best = -1.f;
                for (int i = k; i < NE; ++i) {
                    float2 v = Ms[i][k];
                    float m = v.x * v.x + v.y * v.y;
                    if (m > best) { best = m; piv = i; }
                }
                s_piv = piv;
            }
            __syncthreads();
            const int piv = s_piv;
            if (piv != k) {
                for (int j = tid; j < NE; j += NTHREADS) {
                    float2 t = Ms[k][j]; Ms[k][j] = Ms[piv][j]; Ms[piv][j] = t;
                }
            }
            __syncthreads();
            if (tid == 0) {
                if (piv != k) s_nsw++;
                float2 u = Ms[k][k];
                double m2 = (double)u.x * (double)u.x + (double)u.y * (double)u.y;
                s_lre += 0.5 * log(m2 + 1e-300);
                s_lim += atan2((double)u.y, (double)u.x);
                float m2f = u.x * u.x + u.y * u.y + 1e-30f;
                s_inv = make_float2(u.x / m2f, -u.y / m2f);
            }
            __syncthreads();
            const float2 inv = s_inv;
            for (int i = k + 1 + tid; i < NE; i += NTHREADS)
                Ms[i][k] = cmulf(Ms[i][k], inv);
            __syncthreads();
            // trailing update restricted to current panel columns
            const int nc = (kbase + NB) - (k + 1);
            if (nc > 0) {
                const int nr = NE - (k + 1);
                for (int idx = tid; idx < nr * nc; idx += NTHREADS) {
                    const int i = k + 1 + idx / nc;
                    const int j = k + 1 + idx % nc;
                    float2 l = Ms[i][k], u = Ms[k][j], cc = Ms[i][j];
                    Ms[i][j] = make_float2(cc.x - (l.x * u.x - l.y * u.y),
                                           cc.y - (l.x * u.y + l.y * u.x));
                }
            }
            __syncthreads();
        }

        if (kb < 3) {
            // ---- TRSM: U block-row = L11^{-1} * A12 (unit lower), one column/thread
            const int cbeg  = kbase + NB;
            const int ncols = NE - cbeg;
            if (tid < ncols) {
                const int j = cbeg + tid;
                float2 x[NB];
                for (int ii = 0; ii < NB; ++ii) {
                    float2 a = Ms[kbase + ii][j];
                    for (int m = 0; m < ii; ++m) {
                        float2 l = Ms[kbase + ii][kbase + m];
                        a.x -= l.x * x[m].x - l.y * x[m].y;
                        a.y -= l.x * x[m].y + l.y * x[m].x;
                    }
                    x[ii] = a;
                }
                for (int ii = 0; ii < NB; ++ii) Ms[kbase + ii][j] = x[ii];
            }
            __syncthreads();

            // ---- Schur update C -= L*U via V_WMMA_F32_16X16X4_F32.
            // Complex GEMM: Cre += (-Lre)Bre + (Lim)Bim ; Cim += (-Lre)Bim + (-Lim)Bre
            const int nrem  = 3 - kb;
            const int wave  = tid >> 5;
            const int lane  = tid & 31;
            const int rlo   = lane & 15;
            const int khalf = (lane >> 4) << 1;   // K split: lanes 0-15 -> K0,1; 16-31 -> K2,3
            for (int t = wave; t < nrem * nrem; t += 4) {   // wave-uniform => EXEC all ones
                const int rb = cbeg + NB * (t / nrem);
                const int cb = cbeg + NB * (t % nrem);
                v8f cre{}, cim{};
                #pragma unroll
                for (int v = 0; v < 8; ++v) {
                    const int rr = rb + ((lane < 16) ? v : v + 8);
                    float2 cc = Ms[rr][cb + rlo];
                    cre[v] = cc.x; cim[v] = cc.y;
                }
                #pragma unroll
                for (int kc = 0; kc < 4; ++kc) {            // K=16 as 4 chained x4 WMMAs
                    const int ka = kbase + 4 * kc + khalf;
                    const int ra = rb + rlo;
                    float2 a0 = Ms[ra][ka],      a1 = Ms[ra][ka + 1];
                    float2 b0 = Ms[ka][cb + rlo], b1 = Ms[ka + 1][cb + rlo];
                    v2f are = { -a0.x, -a1.x };
                    v2f aim = { -a0.y, -a1.y };
                    v2f pim = {  a0.y,  a1.y };
                    v2f bre = {  b0.x,  b1.x };
                    v2f bim = {  b0.y,  b1.y };
                    cre = __builtin_amdgcn_wmma_f32_16x16x4_f32(false, are, false, bre, (short)0, cre, false, false);
                    cre = __builtin_amdgcn_wmma_f32_16x16x4_f32(false, pim, false, bim, (short)0, cre, false, false);
                    cim = __builtin_amdgcn_wmma_f32_16x16x4_f32(false, are, false, bim, (short)0, cim, false, false);
                    cim = __builtin_amdgcn_wmma_f32_16x16x4_f32(false, aim, false, bre, (short)0, cim, false, false);
                }
                #pragma unroll
                for (int v = 0; v < 8; ++v) {
                    const int rr = rb + ((lane < 16) ? v : v + 8);
                    Ms[rr][cb + rlo] = make_float2(cre[v], cim[v]);
                }
            }
            __syncthreads();
        }
    }

    // ---- logPf = 0.5*logdet(M) + n^2*log(1j), n=32 -> + i*512*pi
    __syncthreads();
    if (tid == 0) {
        const double PI = 3.14159265358979323846;
        const double ldr = s_lre;
        const double ldi = s_lim + PI * (double)s_nsw;   // each row swap: det *= -1
        outg[2 * (size_t)b + 0] = 0.5 * ldr;
        outg[2 * (size_t)b + 1] = 0.5 * ldi + 512.0 * PI;
    }
}

extern "C" void kernel_launch(void* const* d_in, const int* in_sizes, int n_in,
                              void* d_out, int out_size, void* d_ws, size_t ws_size,
                              hipStream_t stream) {
    const long long* y = (const long long*)d_in[0];   // int64 indices, [1024,64]
    const double*    F = (const double*)d_in[1];      // complex128 as interleaved doubles
    double*          o = (double*)d_out;              // complex128 out, interleaved
    (void)in_sizes; (void)n_in; (void)out_size; (void)d_ws; (void)ws_size;
    pfaffian_logdet_kernel<<<dim3(BATCH_N), dim3(NTHREADS), 0, stream>>>(y, F, o);
}